// CrossAttentionMemory_46110768890110
// MI455X (gfx1250) — compile-verified
//
#include <hip/hip_runtime.h>
#include <math.h>

#define DIMN   2048
#define NB     8
#define NM     16384
#define NCHUNK 64
#define CHUNK  (NM / NCHUNK)   // 256 rows per block
#define TPB    256
#define RECF   2052            // floats per partial record (16B-aligned stride)

typedef __attribute__((ext_vector_type(2))) float v2f;
typedef __attribute__((ext_vector_type(8))) float v8f;

__device__ __forceinline__ v2f ld2(const float* p) {
  return *(const v2f*)p;   // 8B-aligned (even float offsets everywhere below)
}

// ---------------------------------------------------------------------------
// Prologue: xpad[16x2048] = x[8x2048] zero-padded to 16 rows.
// Removes the divergent zero-pad branch from the WMMA inner loop.
// ---------------------------------------------------------------------------
__global__ void pad_x(const float* __restrict__ x, float* __restrict__ xpad) {
  const int i = blockIdx.x * TPB + threadIdx.x;        // 0 .. 16*2048-1
  xpad[i] = (i < NB * DIMN) ? x[i] : 0.f;
}

// ---------------------------------------------------------------------------
// Kernel A: xq[16x2048] = xpad @ wq^T     (v_wmma_f32_16x16x4_f32)
// One wave per 16-wide output tile; two independent accumulators to break the
// WMMA D->C serial chain. B source is wq transposed -> contiguous float2/lane.
// ---------------------------------------------------------------------------
__global__ void proj_q_wmma(const float* __restrict__ xpad, // [16 x 2048]
                            const float* __restrict__ wq,   // [2048 x 2048] [h][d]
                            float* __restrict__ xq)         // [16 x 2048]
{
  const int lane = threadIdx.x & 31;
  const int half = lane >> 4;          // 0: K pair {0,1}, 1: K pair {2,3}
  const int row  = lane & 15;
  const int n0   = blockIdx.x * 16;
  v8f c0 = {0.f,0.f,0.f,0.f,0.f,0.f,0.f,0.f};
  v8f c1 = {0.f,0.f,0.f,0.f,0.f,0.f,0.f,0.f};

  const float* ab = xpad + (size_t)row * DIMN + 2 * half;
  const float* bb = wq + (size_t)(n0 + row) * DIMN + 2 * half;  // B[k][n] = wq[n][k]

  for (int k = 0; k < DIMN; k += 8) {
    const v2f a0 = ld2(ab + k);
    const v2f b0 = ld2(bb + k);
    const v2f a1 = ld2(ab + k + 4);
    const v2f b1 = ld2(bb + k + 4);
    c0 = __builtin_amdgcn_wmma_f32_16x16x4_f32(false, a0, false, b0, (short)0, c0, false, false);
    c1 = __builtin_amdgcn_wmma_f32_16x16x4_f32(false, a1, false, b1, (short)0, c1, false, false);
  }
#pragma unroll
  for (int v = 0; v < 8; ++v) {
    const int m = v + 8 * half;        // C layout: VGPR v -> rows v / v+8
    xq[(size_t)m * DIMN + n0 + row] = c0[v] + c1[v];
  }
}

// ---------------------------------------------------------------------------
// Kernel B: qt[16x2048] = xq @ wk        (v_wmma_f32_16x16x4_f32)
// wk is already row-major K x N; B fragment = two row-strided scalar loads.
// ---------------------------------------------------------------------------
__global__ void proj_k_wmma(const float* __restrict__ xq,  // [16 x 2048]
                            const float* __restrict__ wk,  // [2048 x 2048] = B[k][n]
                            float* __restrict__ qt)        // [16 x 2048]
{
  const int lane = threadIdx.x & 31;
  const int half = lane >> 4;
  const int row  = lane & 15;
  const int n0   = blockIdx.x * 16;
  v8f c0 = {0.f,0.f,0.f,0.f,0.f,0.f,0.f,0.f};
  v8f c1 = {0.f,0.f,0.f,0.f,0.f,0.f,0.f,0.f};

  const float* ab = xq + (size_t)row * DIMN + 2 * half;
  const float* bc = wk + (size_t)(2 * half) * DIMN + n0 + row;

  for (int k = 0; k < DIMN; k += 8) {
    const v2f a0 = ld2(ab + k);
    const v2f a1 = ld2(ab + k + 4);
    v2f b0, b1;
    const float* bp = bc + (size_t)k * DIMN;
    b0.x = bp[0];
    b0.y = bp[DIMN];
    b1.x = bp[4 * DIMN];
    b1.y = bp[5 * DIMN];
    c0 = __builtin_amdgcn_wmma_f32_16x16x4_f32(false, a0, false, b0, (short)0, c0, false, false);
    c1 = __builtin_amdgcn_wmma_f32_16x16x4_f32(false, a1, false, b1, (short)0, c1, false, false);
  }
#pragma unroll
  for (int v = 0; v < 8; ++v) {
    const int m = v + 8 * half;
    qt[(size_t)m * DIMN + n0 + row] = c0[v] + c1[v];
  }
}

// ---------------------------------------------------------------------------
// Kernel C: fused streaming pass over `memory` with online softmax.
// Block = (batch b, chunk of 256 rows); thread owns 8 columns (2x float4).
// Two rows per iteration -> one __syncthreads round per 2 rows.
// Emits per-block partials {acc[2048], max, sumexp}.
// ---------------------------------------------------------------------------
__global__ void attn_stream(const float* __restrict__ mem,  // [8 x 16384 x 2048]
                            const float* __restrict__ qt,   // [16 x 2048], row b
                            float* __restrict__ part)
{
  const int b     = blockIdx.y;
  const int chunk = blockIdx.x;
  const int t     = threadIdx.x;
  const float scale = 0.022097086912079612f;   // 1/sqrt(2048)
  __shared__ float red[2][TPB / 32];

  const float* qp = qt + (size_t)b * DIMN + t * 8;
  const float4 q0 = *(const float4*)qp;
  const float4 q1 = *(const float4*)(qp + 4);

  float4 a0 = {0.f,0.f,0.f,0.f}, a1 = {0.f,0.f,0.f,0.f};
  float mx = -INFINITY, sm = 0.f;

  const float* base = mem + ((size_t)b * NM + (size_t)chunk * CHUNK) * DIMN + t * 8;

  for (int r = 0; r < CHUNK; r += 2) {
    const float* rpA = base + (size_t)r * DIMN;
    const float* rpB = rpA + DIMN;
    if (r + 8 < CHUNK) {
      __builtin_prefetch(rpA + 8 * DIMN, 0, 1);
      __builtin_prefetch(rpB + 8 * DIMN, 0, 1);
    }
    const float4 rA0 = *(const float4*)rpA;
    const float4 rA1 = *(const float4*)(rpA + 4);
    const float4 rB0 = *(const float4*)rpB;
    const float4 rB1 = *(const float4*)(rpB + 4);

    float pA = rA0.x*q0.x + rA0.y*q0.y + rA0.z*q0.z + rA0.w*q0.w
             + rA1.x*q1.x + rA1.y*q1.y + rA1.z*q1.z + rA1.w*q1.w;
    float pB = rB0.x*q0.x + rB0.y*q0.y + rB0.z*q0.z + rB0.w*q0.w
             + rB1.x*q1.x + rB1.y*q1.y + rB1.z*q1.z + rB1.w*q1.w;
#pragma unroll
    for (int off = 16; off > 0; off >>= 1) {
      pA += __shfl_xor(pA, off, 32);
      pB += __shfl_xor(pB, off, 32);
    }
    if ((t & 31) == 0) { red[0][t >> 5] = pA; red[1][t >> 5] = pB; }
    __syncthreads();
    float sA = 0.f, sB = 0.f;
#pragma unroll
    for (int w = 0; w < TPB / 32; ++w) { sA += red[0][w]; sB += red[1][w]; }
    __syncthreads();
    sA *= scale;
    sB *= scale;

    // Online softmax; sA/sB/mx are block-uniform -> uniform branches.
    if (sA > mx) {
      const float f = __expf(mx - sA);     // exp(-inf)=0 handles first row
      sm = sm * f + 1.f;
      a0.x = a0.x*f + rA0.x; a0.y = a0.y*f + rA0.y; a0.z = a0.z*f + rA0.z; a0.w = a0.w*f + rA0.w;
      a1.x = a1.x*f + rA1.x; a1.y = a1.y*f + rA1.y; a1.z = a1.z*f + rA1.z; a1.w = a1.w*f + rA1.w;
      mx = sA;
    } else {
      const float e = __expf(sA - mx);
      sm += e;
      a0.x += e*rA0.x; a0.y += e*rA0.y; a0.z += e*rA0.z; a0.w += e*rA0.w;
      a1.x += e*rA1.x; a1.y += e*rA1.y; a1.z += e*rA1.z; a1.w += e*rA1.w;
    }
    if (sB > mx) {
      const float f = __expf(mx - sB);
      sm = sm * f + 1.f;
      a0.x = a0.x*f + rB0.x; a0.y = a0.y*f + rB0.y; a0.z = a0.z*f + rB0.z; a0.w = a0.w*f + rB0.w;
      a1.x = a1.x*f + rB1.x; a1.y = a1.y*f + rB1.y; a1.z = a1.z*f + rB1.z; a1.w = a1.w*f + rB1.w;
      mx = sB;
    } else {
      const float e = __expf(sB - mx);
      sm += e;
      a0.x += e*rB0.x; a0.y += e*rB0.y; a0.z += e*rB0.z; a0.w += e*rB0.w;
      a1.x += e*rB1.x; a1.y += e*rB1.y; a1.z += e*rB1.z; a1.w += e*rB1.w;
    }
  }

  float* rec = part + (size_t)(b * NCHUNK + chunk) * RECF;
  *(float4*)(rec + t * 8)     = a0;
  *(float4*)(rec + t * 8 + 4) = a1;
  if (t == 0) { rec[2048] = mx; rec[2049] = sm; }
}

// ---------------------------------------------------------------------------
// Kernel D: combine per-chunk partials -> softmax-normalized output [8 x 2048].
// ---------------------------------------------------------------------------
__global__ void attn_reduce(const float* __restrict__ part,
                            float* __restrict__ out)
{
  const int b = blockIdx.x;
  const int t = threadIdx.x;
  const float* base = part + (size_t)b * NCHUNK * RECF;

  float MX = -INFINITY;
  for (int c = 0; c < NCHUNK; ++c) MX = fmaxf(MX, base[(size_t)c * RECF + 2048]);

  float total = 0.f;
  float4 o0 = {0.f,0.f,0.f,0.f}, o1 = {0.f,0.f,0.f,0.f};
  for (int c = 0; c < NCHUNK; ++c) {
    const float* rec = base + (size_t)c * RECF;
    const float f = __expf(rec[2048] - MX);
    total += rec[2049] * f;
    const float4 a0 = *(const float4*)(rec + t * 8);
    const float4 a1 = *(const float4*)(rec + t * 8 + 4);
    o0.x += f*a0.x; o0.y += f*a0.y; o0.z += f*a0.z; o0.w += f*a0.w;
    o1.x += f*a1.x; o1.y += f*a1.y; o1.z += f*a1.z; o1.w += f*a1.w;
  }
  const float inv = 1.f / total;
  float* op = out + (size_t)b * DIMN + t * 8;
  float4 w0 = {o0.x*inv, o0.y*inv, o0.z*inv, o0.w*inv};
  float4 w1 = {o1.x*inv, o1.y*inv, o1.z*inv, o1.w*inv};
  *(float4*)op       = w0;
  *(float4*)(op + 4) = w1;
}

// ---------------------------------------------------------------------------
extern "C" void kernel_launch(void* const* d_in, const int* in_sizes, int n_in,
                              void* d_out, int out_size, void* d_ws, size_t ws_size,
                              hipStream_t stream) {
  const float* mem = (const float*)d_in[0];   // memory [8,16384,2048]
  const float* x   = (const float*)d_in[1];   // inputs [8,1,2048]
  const float* wq  = (const float*)d_in[2];   // [2048,2048]
  const float* wk  = (const float*)d_in[3];   // [2048,2048]
  float* out = (float*)d_out;                 // [8,1,2048]

  float* ws   = (float*)d_ws;
  float* xpad = ws;                           // 16*2048 floats
  float* xq   = ws + 16 * DIMN;               // 16*2048 floats
  float* qt   = ws + 32 * DIMN;               // 16*2048 floats
  float* part = ws + 48 * DIMN;               // 512 * RECF floats (~4.2 MB)

  pad_x<<<(16 * DIMN) / TPB, TPB, 0, stream>>>(x, xpad);
  proj_q_wmma<<<DIMN / 16, 32, 0, stream>>>(xpad, wq, xq);
  proj_k_wmma<<<DIMN / 16, 32, 0, stream>>>(xq, wk, qt);
  dim3 grid(NCHUNK, NB);
  attn_stream<<<grid, TPB, 0, stream>>>(mem, qt, part);
  attn_reduce<<<NB, TPB, 0, stream>>>(part, out);
}